// EnhancedGraphNeuralNetwork_4810363372589
// MI455X (gfx1250) — compile-verified
//
#include <hip/hip_runtime.h>
#include <cstdint>
#include <cstddef>

#define F_HID 128
#define N_CLS 40
#define BN_EPS 1e-5f

typedef __attribute__((ext_vector_type(2))) float v2f;
typedef __attribute__((ext_vector_type(4))) float v4f;
typedef __attribute__((ext_vector_type(8))) float v8f;

// ---------------- degree / dinv ----------------
__global__ void init_deg_kernel(float* __restrict__ deg, int n) {
    int i = blockIdx.x * blockDim.x + threadIdx.x;
    if (i < n) deg[i] = 1.0f;                       // self loop
}

__global__ void count_deg_kernel(const int* __restrict__ dst, float* __restrict__ deg, int e) {
    int i = blockIdx.x * blockDim.x + threadIdx.x;
    if (i < e) unsafeAtomicAdd(&deg[dst[i]], 1.0f);
}

__global__ void rsqrt_kernel(float* __restrict__ d, int n) {
    int i = blockIdx.x * blockDim.x + threadIdx.x;
    if (i < n) d[i] = rsqrtf(d[i]);
}

// ---------------- WMMA fp32 GEMM: T[n,WC] = X[n,128] @ W[128,WC] ----------------
// One wave computes one 16x16 tile via 32x V_WMMA_F32_16X16X4_F32.
// A 16x4 layout (ISA 7.12.2): VGPR0 = K{0 | 2}, VGPR1 = K{1 | 3} for lane halves.
// C/D layout: VGPR i: lanes 0-15 -> row i, lanes 16-31 -> row 8+i, col = lane%16.
// For WC not a multiple of 16 we CLAMP the B column instead of predicating the
// load: the clamped lanes feed only accumulator columns that are never stored,
// so the inner loop stays branchless (no exec-mask churn around the WMMAs).
template <int WC>
__global__ __launch_bounds__(256) void gemm_wmma_kernel(const float* __restrict__ X,
                                                        const float* __restrict__ W,
                                                        float* __restrict__ T, int nrows) {
    const int K = F_HID;
    int wave = threadIdx.x >> 5;
    int lane = threadIdx.x & 31;
    int half = lane >> 4;        // 0: lanes 0-15, 1: lanes 16-31
    int l    = lane & 15;
    int row0 = blockIdx.x * 16;
    int col  = wave * 16 + l;

    int colc = col;
    if (WC % 16 != 0) colc = (col < WC) ? col : (WC - 1);   // branchless clamp (v_min)

    const float* Ap = X + (size_t)(row0 + l) * K + half * 2;     // A fragment base
    const float* Bp = W + (size_t)(half * 2) * WC + colc;        // B fragment base

    v8f acc = {0.f, 0.f, 0.f, 0.f, 0.f, 0.f, 0.f, 0.f};
    #pragma unroll 8
    for (int k0 = 0; k0 < K; k0 += 4) {
        v2f a = *(const v2f*)(Ap + k0);                // K = k0+half*2, +1 (b64 load)
        v2f b;
        b.x = Bp[(size_t)k0 * WC];
        b.y = Bp[(size_t)k0 * WC + WC];
        acc = __builtin_amdgcn_wmma_f32_16x16x4_f32(false, a, false, b,
                                                    (short)0, acc, false, false);
    }
    if ((WC % 16 == 0 || col < WC) && row0 + 15 < nrows) {
        #pragma unroll
        for (int i = 0; i < 8; ++i) {
            int row = row0 + half * 8 + i;
            T[(size_t)row * WC + col] = acc[i];
        }
    }
}

// ---------------- aggregation ----------------
// A[i,:] = T[i,:] * dinv[i]^2   (self-loop term doubles as initialization)
template <int F>
__global__ void selfloop_init_kernel(const float* __restrict__ T, const float* __restrict__ dinv,
                                     float* __restrict__ A, int n) {
    int idx = blockIdx.x * blockDim.x + threadIdx.x;
    if (idx < n * F) {
        float di = dinv[idx / F];
        A[idx] = T[idx] * di * di;
    }
}

// A[dst,:] += T[src,:] * dinv[src]*dinv[dst], 4 floats per thread
template <int F>
__global__ void aggregate_kernel(const float* __restrict__ T, const int* __restrict__ src,
                                 const int* __restrict__ dst, const float* __restrict__ dinv,
                                 float* __restrict__ A, int e) {
    int idx = blockIdx.x * blockDim.x + threadIdx.x;
    constexpr int CPE = F >> 2;                      // float4 chunks per edge
    int ed = idx / CPE;                              // const divisor -> mul/shift
    int c  = idx - ed * CPE;
    if (ed >= e) return;
    int s = src[ed], d = dst[ed];
    float coef = dinv[s] * dinv[d];
    const v4f v = *(const v4f*)(T + (size_t)s * F + c * 4);
    float* o = A + (size_t)d * F + c * 4;
    unsafeAtomicAdd(o + 0, v.x * coef);
    unsafeAtomicAdd(o + 1, v.y * coef);
    unsafeAtomicAdd(o + 2, v.z * coef);
    unsafeAtomicAdd(o + 3, v.w * coef);
}

// ---------------- batch-norm ----------------
__global__ void zero_stats_kernel(float* __restrict__ stats) {
    stats[threadIdx.x] = 0.0f;                      // launched with 2*F_HID threads
}

__global__ __launch_bounds__(128) void colstats_kernel(const float* __restrict__ A,
                                                       float* __restrict__ stats, int n) {
    int col = threadIdx.x;
    int r0 = blockIdx.x * 256;
    int r1 = r0 + 256; if (r1 > n) r1 = n;
    float s = 0.f, sq = 0.f;
    for (int r = r0; r < r1; ++r) {
        float v = A[(size_t)r * F_HID + col];
        s += v; sq += v * v;
    }
    unsafeAtomicAdd(&stats[col], s);
    unsafeAtomicAdd(&stats[F_HID + col], sq);
}

// H = relu( (A - mean)*rsqrt(var+eps)*g + be (+ res) )
__global__ void bn_relu_kernel(const float* __restrict__ A, const float* __restrict__ stats,
                               const float* __restrict__ g, const float* __restrict__ be,
                               const float* __restrict__ res, float* __restrict__ H, int n) {
    int idx = blockIdx.x * blockDim.x + threadIdx.x;
    if (idx >= n * F_HID) return;
    int col = idx & (F_HID - 1);
    float invN = 1.0f / (float)n;
    float mean = stats[col] * invN;
    float var  = stats[F_HID + col] * invN - mean * mean;
    float v = (A[idx] - mean) * rsqrtf(var + BN_EPS) * g[col] + be[col];
    if (res) v += res[idx];
    H[idx] = fmaxf(v, 0.0f);
}

// ---------------- log-softmax over 40 classes, one wave32 per row ----------------
__global__ __launch_bounds__(128) void logsoftmax_kernel(const float* __restrict__ A,
                                                         const float* __restrict__ b3,
                                                         float* __restrict__ out, int n) {
    int wave = threadIdx.x >> 5;
    int lane = threadIdx.x & 31;
    int row = blockIdx.x * 4 + wave;
    if (row >= n) return;
    const float* in = A + (size_t)row * N_CLS;
    bool hi = (lane + 32) < N_CLS;
    float v0 = in[lane] + b3[lane];
    float v1 = hi ? (in[lane + 32] + b3[lane + 32]) : -3.402823466e38f;
    float m = fmaxf(v0, v1);
    #pragma unroll
    for (int off = 16; off > 0; off >>= 1) m = fmaxf(m, __shfl_xor(m, off, 32));
    float s = __expf(v0 - m) + (hi ? __expf(v1 - m) : 0.0f);
    #pragma unroll
    for (int off = 16; off > 0; off >>= 1) s += __shfl_xor(s, off, 32);
    float lse = m + __logf(s);
    out[(size_t)row * N_CLS + lane] = v0 - lse;
    if (hi) out[(size_t)row * N_CLS + lane + 32] = v1 - lse;
}

extern "C" void kernel_launch(void* const* d_in, const int* in_sizes, int n_in,
                              void* d_out, int out_size, void* d_ws, size_t ws_size,
                              hipStream_t stream) {
    const float* x  = (const float*)d_in[0];
    const int*   ei = (const int*)d_in[1];
    const float* W1 = (const float*)d_in[2];
    // b1 (d_in[3]) cancels inside BN; skipped
    const float* g1  = (const float*)d_in[4];
    const float* be1 = (const float*)d_in[5];
    const float* W2  = (const float*)d_in[6];
    // b2 (d_in[7]) cancels inside BN; skipped
    const float* g2  = (const float*)d_in[8];
    const float* be2 = (const float*)d_in[9];
    const float* W3  = (const float*)d_in[10];
    const float* b3  = (const float*)d_in[11];
    float* out = (float*)d_out;

    int n = in_sizes[0] / F_HID;      // 100000
    int E = in_sizes[1] / 2;          // 1600000
    const int* src = ei;
    const int* dst = ei + E;

    float* ws    = (float*)d_ws;
    float* dinv  = ws;                            // n
    float* T     = dinv + n;                      // n*128
    float* A     = T + (size_t)n * F_HID;         // n*128
    float* H     = A + (size_t)n * F_HID;         // n*128
    float* stats = H + (size_t)n * F_HID;         // 2*128
    float* T3    = stats + 2 * F_HID;             // n*40
    float* A3    = T3 + (size_t)n * N_CLS;        // n*40

    int gN   = (n + 255) / 256;
    int gE   = (E + 255) / 256;
    int gNF  = (int)(((size_t)n * F_HID + 255) / 256);
    int gNC  = (int)(((size_t)n * N_CLS + 255) / 256);
    int gAgg = (int)(((size_t)E * (F_HID / 4) + 255) / 256);
    int gAg3 = (int)(((size_t)E * (N_CLS / 4) + 255) / 256);
    int rowTiles = (n + 15) / 16;

    // degree / dinv
    init_deg_kernel<<<gN, 256, 0, stream>>>(dinv, n);
    count_deg_kernel<<<gE, 256, 0, stream>>>(dst, dinv, E);
    rsqrt_kernel<<<gN, 256, 0, stream>>>(dinv, n);

    // ---- layer 1: conv + BN + residual + relu ----
    gemm_wmma_kernel<F_HID><<<rowTiles, 256, 0, stream>>>(x, W1, T, n);
    selfloop_init_kernel<F_HID><<<gNF, 256, 0, stream>>>(T, dinv, A, n);
    aggregate_kernel<F_HID><<<gAgg, 256, 0, stream>>>(T, src, dst, dinv, A, E);
    zero_stats_kernel<<<1, 2 * F_HID, 0, stream>>>(stats);
    colstats_kernel<<<gN, 128, 0, stream>>>(A, stats, n);
    bn_relu_kernel<<<gNF, 256, 0, stream>>>(A, stats, g1, be1, x, H, n);

    // ---- layer 2: conv + BN + relu ----
    gemm_wmma_kernel<F_HID><<<rowTiles, 256, 0, stream>>>(H, W2, T, n);
    selfloop_init_kernel<F_HID><<<gNF, 256, 0, stream>>>(T, dinv, A, n);
    aggregate_kernel<F_HID><<<gAgg, 256, 0, stream>>>(T, src, dst, dinv, A, E);
    zero_stats_kernel<<<1, 2 * F_HID, 0, stream>>>(stats);
    colstats_kernel<<<gN, 128, 0, stream>>>(A, stats, n);
    bn_relu_kernel<<<gNF, 256, 0, stream>>>(A, stats, g2, be2, nullptr, H, n);

    // ---- layer 3: conv + log_softmax ----
    gemm_wmma_kernel<N_CLS><<<rowTiles, 96, 0, stream>>>(H, W3, T3, n);
    selfloop_init_kernel<N_CLS><<<gNC, 256, 0, stream>>>(T3, dinv, A3, n);
    aggregate_kernel<N_CLS><<<gAg3, 256, 0, stream>>>(T3, src, dst, dinv, A3, E);
    logsoftmax_kernel<<<(n + 3) / 4, 128, 0, stream>>>(A3, b3, out, n);
}